// CustomLossCNN1DFast_11364483465834
// MI455X (gfx1250) — compile-verified
//
#include <hip/hip_runtime.h>

// ---------------------------------------------------------------------------
// CDNA5 / gfx1250 implementation.
//   loss = mean_b [ T2 - 2a*TA + a^2*AA ],  a = TP/PA
//   T2=<t,t> TP=<t,p> TA=<t,A p> PA=<p,A p> AA=<A p,A p>   (per batch row b)
// Pipeline: transpose p -> scatter COO into aT -> WMMA Gram diagonals -> final.
// ---------------------------------------------------------------------------

typedef __attribute__((ext_vector_type(2))) float v2f;
typedef __attribute__((ext_vector_type(4))) float v4f;
typedef __attribute__((ext_vector_type(8))) float v8f;

#define BS 32
#define CHUNK_COLS 4096

// D = A(16x4) * B(4x16) + C ; f32 everywhere.
// Args: (neg_a, A, neg_b, B, c_mod, C, reuse_a, reuse_b)  [probe-confirmed]
static __device__ __forceinline__ v8f wmma4(v2f a, v2f b, v8f c) {
  return __builtin_amdgcn_wmma_f32_16x16x4_f32(false, a, false, b, (short)0, c,
                                               false, false);
}

// ---------------------------------------------------------------------------
// Kernel 1: transpose y_pred (BS,N) row-major -> pT (N,BS).
// Reads coalesced across lanes; each thread owns a 128B output row -> 8x b128.
// ---------------------------------------------------------------------------
__global__ void k_transpose(const float* __restrict__ yp,
                            float* __restrict__ pT, int n_total) {
  int n = blockIdx.x * blockDim.x + threadIdx.x;
  if (n >= n_total) return;
  float buf[BS];
#pragma unroll
  for (int b = 0; b < BS; ++b) buf[b] = yp[(size_t)b * n_total + n];
  v4f* dst = (v4f*)(pT + (size_t)n * BS);
#pragma unroll
  for (int q = 0; q < BS / 4; ++q) {
    v4f v = {buf[4 * q], buf[4 * q + 1], buf[4 * q + 2], buf[4 * q + 3]};
    dst[q] = v;
  }
}

// ---------------------------------------------------------------------------
// Kernel 2: zero aT and the dot-product accumulators.
// ---------------------------------------------------------------------------
__global__ void k_zero(float* __restrict__ p, size_t count) {
  size_t i = (size_t)blockIdx.x * blockDim.x + threadIdx.x;
  if (i < count) p[i] = 0.0f;
}

// ---------------------------------------------------------------------------
// Kernel 3: COO scatter. One wave per nonzero, lane = batch index.
// e is wave-uniform: force it scalar so vals/rows/cols become SMEM loads.
// Gather pT[col,*] (128B contiguous) and atomic-add into aT[row,*]
// (128B contiguous). pT/aT together = 67MB -> resident in 192MB L2.
// ---------------------------------------------------------------------------
__global__ void k_scatter(const float* __restrict__ vals,
                          const int* __restrict__ rows,
                          const int* __restrict__ cols,
                          const float* __restrict__ pT,
                          float* __restrict__ aT, int nnz) {
  int waveId = __builtin_amdgcn_readfirstlane(
      (int)((blockIdx.x * blockDim.x + threadIdx.x) >> 5));
  int b = threadIdx.x & 31;
  int e = waveId;
  if (e >= nnz) return;  // wave-uniform
  float v = vals[e];
  int c = cols[e];
  int r = rows[e];
  atomicAdd(&aT[(size_t)r * BS + b], v * pT[(size_t)c * BS + b]);
}

// ---------------------------------------------------------------------------
// Kernel 4: five Gram diagonals via V_WMMA_F32_16X16X4_F32.
// One wave handles one batch-half g (16 rows) over one column chunk.
// Per-lane layout (ISA 7.12.2, 32-bit 16x4 A): lane l, h=l>>4, m=l&15 holds
// X[m, 2h] and X[m, 2h+1]  -> one float2 load feeds BOTH the A and B slots
// (symmetric K-striping of the 4x16 B operand).
// C diagonal (m,m): VGPR m lane m for m<8; VGPR m-8 lane 16+m for m>=8.
// wave/chunk forced through readfirstlane so the loop is scalar-controlled
// and EXEC is statically all-ones around every WMMA.
// ---------------------------------------------------------------------------
__global__ void k_dots_wmma(const float* __restrict__ y_pred,
                            const float* __restrict__ y_true,
                            const float* __restrict__ aT,
                            float* __restrict__ dots, int N, int nChunks) {
  const int lane = threadIdx.x & 31;
  const int wave = __builtin_amdgcn_readfirstlane(
      (int)(blockIdx.x * (blockDim.x >> 5) + (threadIdx.x >> 5)));
  const int totalWaves = 2 * nChunks;
  if (wave >= totalWaves) return;  // scalar branch
  const int g = wave & 1;          // batch half: rows [16g, 16g+16)
  const int chunk = wave >> 1;

  const int half = lane >> 4;
  const int m = lane & 15;
  const int b = g * 16 + m;
  const size_t rowBase = (size_t)b * N;

  v8f cT2 = {}, cTP = {}, cTA = {}, cPA = {}, cAA = {};

  const int c0 = chunk * CHUNK_COLS;                         // SGPR
  const int c1 = (c0 + CHUNK_COLS < N) ? (c0 + CHUNK_COLS)   // SGPR
                                       : N;
#pragma unroll 2
  for (int c = c0; c < c1; c += 4) {
    const size_t col = (size_t)c + 2 * half;  // this lane's K pair
    v2f t = *(const v2f*)(y_true + rowBase + col);
    v2f p = *(const v2f*)(y_pred + rowBase + col);
    v2f a;
    a.x = aT[col * BS + b];
    a.y = aT[(col + 1) * BS + b];
    cT2 = wmma4(t, t, cT2);
    cTP = wmma4(t, p, cTP);
    cTA = wmma4(t, a, cTA);
    cPA = wmma4(p, a, cPA);
    cAA = wmma4(a, a, cAA);
  }

  // Diagonal extraction: lane r in {0..7} owns diag m=r (VGPR r); lane 24+r
  // owns diag m=8+r (VGPR r). Select via compile-time unrolled cndmask tree,
  // then 5 atomics behind a single predicate.
  const int r = lane & 7;
  float d0 = 0.f, d1 = 0.f, d2 = 0.f, d3 = 0.f, d4 = 0.f;
#pragma unroll
  for (int j = 0; j < 8; ++j) {
    if (r == j) {
      d0 = cT2[j]; d1 = cTP[j]; d2 = cTA[j]; d3 = cPA[j]; d4 = cAA[j];
    }
  }
  const bool emit = (lane < 8) || (lane >= 24);
  const int bIdx = g * 16 + ((lane < 8) ? lane : (lane - 16));
  if (emit) {
    atomicAdd(&dots[0 * BS + bIdx], d0);
    atomicAdd(&dots[1 * BS + bIdx], d1);
    atomicAdd(&dots[2 * BS + bIdx], d2);
    atomicAdd(&dots[3 * BS + bIdx], d3);
    atomicAdd(&dots[4 * BS + bIdx], d4);
  }
}

// ---------------------------------------------------------------------------
// Kernel 5: one wave, lane = batch. alpha, expanded loss, wave reduce, mean.
// ---------------------------------------------------------------------------
__global__ void k_final(const float* __restrict__ dots,
                        float* __restrict__ out) {
  int b = threadIdx.x;
  float T2 = dots[0 * BS + b];
  float TP = dots[1 * BS + b];
  float TA = dots[2 * BS + b];
  float PA = dots[3 * BS + b];
  float AA = dots[4 * BS + b];
  float alpha = TP / PA;
  float loss = T2 - 2.0f * alpha * TA + alpha * alpha * AA;
#pragma unroll
  for (int off = 16; off > 0; off >>= 1) loss += __shfl_down(loss, off);
  if (b == 0) out[0] = loss * (1.0f / (float)BS);
}

// ---------------------------------------------------------------------------
extern "C" void kernel_launch(void* const* d_in, const int* in_sizes, int n_in,
                              void* d_out, int out_size, void* d_ws,
                              size_t ws_size, hipStream_t stream) {
  const float* y_pred = (const float*)d_in[0];
  const float* y_true = (const float*)d_in[1];
  const float* A_vals = (const float*)d_in[2];
  const int* A_rows = (const int*)d_in[3];
  const int* A_cols = (const int*)d_in[4];
  const int N = in_sizes[0] / BS;
  const int nnz = in_sizes[2];

  // Workspace layout: pT (N*32 f32) | aT (N*32 f32) | dots (5*32 f32)
  float* pT = (float*)d_ws;
  float* aT = pT + (size_t)N * BS;
  float* dots = aT + (size_t)N * BS;

  // 1) transpose y_pred -> pT
  k_transpose<<<(N + 255) / 256, 256, 0, stream>>>(y_pred, pT, N);

  // 2) zero aT + dots
  size_t zc = (size_t)N * BS + 5 * BS;
  k_zero<<<(unsigned)((zc + 255) / 256), 256, 0, stream>>>(aT, zc);

  // 3) COO scatter (one wave per nonzero)
  long long sthreads = (long long)nnz * BS;
  k_scatter<<<(unsigned)((sthreads + 255) / 256), 256, 0, stream>>>(
      A_vals, A_rows, A_cols, pT, aT, nnz);

  // 4) WMMA Gram diagonals
  int nChunks = (N + CHUNK_COLS - 1) / CHUNK_COLS;
  int waves = 2 * nChunks;
  int blocks = (waves + 3) / 4;  // 4 waves (128 threads) per block
  k_dots_wmma<<<blocks, 128, 0, stream>>>(y_pred, y_true, aT, dots, N,
                                          nChunks);

  // 5) final scalar
  k_final<<<1, 32, 0, stream>>>(dots, (float*)d_out);
}